// SpatialGatingNetwork_50629074486093
// MI455X (gfx1250) — compile-verified
//
#include <hip/hip_runtime.h>
#include <hip/hip_bf16.h>

typedef __attribute__((ext_vector_type(2))) float v2f;
typedef __attribute__((ext_vector_type(4))) float v4f;
typedef __attribute__((ext_vector_type(8))) float v8f;

#define KSEL 16
#define DIM  512
#define HID  64

// ---------------------------------------------------------------------------
// Merge one lane-resident top-16 list across the wave: 16 rounds of
// (unrolled local scan -> wave fmin reduce -> lowest-tied-lane removal).
// Returns mean of sqrt of the 16 global minima.
// ---------------------------------------------------------------------------
__device__ __forceinline__ float merge_top16(float best[KSEL], int lane) {
  float sum = 0.0f;
  for (int t = 0; t < KSEL; ++t) {
    float lmin = best[0]; int li = 0;
    #pragma unroll
    for (int j = 1; j < KSEL; ++j) {
      if (best[j] < lmin) { lmin = best[j]; li = j; }
    }
    float gmin = lmin;
    #pragma unroll
    for (int off = 16; off > 0; off >>= 1)
      gmin = fminf(gmin, __shfl_xor(gmin, off, 32));
    unsigned long long m = __ballot(lmin == gmin);
    int owner = __ffsll(m) - 1;          // lowest tied lane removes its copy
    if (lane == owner) {
      #pragma unroll
      for (int j = 0; j < KSEL; ++j) best[j] = (j == li) ? 3.0e38f : best[j];
    }
    sum += sqrtf(fmaxf(gmin, 0.0f));
  }
  return sum * (1.0f / (float)KSEL);
}

// ---------------------------------------------------------------------------
// Phase 1: per-row top-16 nearest (squared) distances, averaged after sqrt.
// One wave32 per TWO rows: each coordinate load is reused for both query
// points, halving L2 traffic (the 800 KB coordinate array is re-streamed by
// every wave; it is the dominant cost of the whole problem). Candidate lists
// are register-resident; dynamic-index writes are unrolled cndmasks so
// nothing spills to scratch.
// ---------------------------------------------------------------------------
__global__ __launch_bounds__(32) void topk_avg_kernel(
    const float* __restrict__ cur,    // [B,2]
    const float* __restrict__ train,  // [N,2]
    float* __restrict__ avg,          // [B]
    int N)
{
  const int b0   = blockIdx.x * 2;
  const int lane = threadIdx.x;
  const float x0 = cur[2 * b0 + 0], y0 = cur[2 * b0 + 1];
  const float x1 = cur[2 * b0 + 2], y1 = cur[2 * b0 + 3];

  float best0[KSEL], best1[KSEL];
  #pragma unroll
  for (int j = 0; j < KSEL; ++j) { best0[j] = 3.0e38f; best1[j] = 3.0e38f; }
  float bmax0 = 3.0e38f, bmax1 = 3.0e38f;
  int   bi0 = 0, bi1 = 0;

  const v2f* __restrict__ t2 = (const v2f*)train;
  for (int i = lane; i < N; i += 32) {
    v2f t = t2[i];                       // one coalesced b64 load, two rows
    float dx0 = x0 - t.x, dy0 = y0 - t.y;
    float dx1 = x1 - t.x, dy1 = y1 - t.y;
    float d0 = fmaf(dx0, dx0, dy0 * dy0);
    float d1 = fmaf(dx1, dx1, dy1 * dy1);
    if (d0 < bmax0) {                    // rare after warm-up
      #pragma unroll
      for (int j = 0; j < KSEL; ++j) best0[j] = (j == bi0) ? d0 : best0[j];
      bmax0 = best0[0]; bi0 = 0;
      #pragma unroll
      for (int j = 1; j < KSEL; ++j)
        if (best0[j] > bmax0) { bmax0 = best0[j]; bi0 = j; }
    }
    if (d1 < bmax1) {
      #pragma unroll
      for (int j = 0; j < KSEL; ++j) best1[j] = (j == bi1) ? d1 : best1[j];
      bmax1 = best1[0]; bi1 = 0;
      #pragma unroll
      for (int j = 1; j < KSEL; ++j)
        if (best1[j] > bmax1) { bmax1 = best1[j]; bi1 = j; }
    }
  }

  float a0 = merge_top16(best0, lane);
  float a1 = merge_top16(best1, lane);
  if (lane == 0) { avg[b0] = a0; avg[b0 + 1] = a1; }
}

// ---------------------------------------------------------------------------
// One-time repack of W1 rows 0..511 into pair-interleaved layout so a WMMA
// B-fragment is a single coalesced b64 load:
//   Wp[k2*128 + n*2 + j] = W1[(2*k2 + j)*64 + n],  k2 in [0,256), j in {0,1}
// ---------------------------------------------------------------------------
__global__ __launch_bounds__(256) void pack_w1_kernel(
    const float* __restrict__ W1, float* __restrict__ Wp)
{
  const int p  = blockIdx.x * 256 + threadIdx.x;   // 0 .. 32767
  const int k2 = p >> 7;
  const int r  = p & 127;
  const int n  = r >> 1;
  const int j  = r & 1;
  Wp[p] = W1[(2 * k2 + j) * HID + n];
}

// ---------------------------------------------------------------------------
// Phase 2+3: h = relu([feat | avg] @ W1 + b1); out = sigmoid(h @ W2 + b2).
// One block = 16 rows, 4 waves; each wave owns a 16-wide N-tile of the 64
// hidden units and runs a K=512 loop of V_WMMA_F32_16X16X4_F32.
// The 16x512 A tile is staged into LDS with ASYNCcnt-tracked
// global_load_async_to_lds_b128 (no VGPR round trip), fenced with
// s_wait_asynccnt before the workgroup barrier.
// ---------------------------------------------------------------------------
__global__ __launch_bounds__(128) void mlp_wmma_kernel(
    const float* __restrict__ feat,  // [B,512]
    const float* __restrict__ avg,   // [B]
    const float* __restrict__ Wp,    // [512*64] pair-interleaved W1
    const float* __restrict__ W1,    // [513,64] (row 512 = avg column)
    const float* __restrict__ b1,    // [64]
    const float* __restrict__ W2,    // [64]
    const float* __restrict__ b2,    // [1]
    float* __restrict__ out)         // [B]
{
  __shared__ float ldsA[16 * DIM];   // 32 KB A tile
  __shared__ float ldsAvg[16];
  __shared__ float ldsH[16 * HID];   // 4 KB hidden tile

  const int tid  = threadIdx.x;
  const int row0 = blockIdx.x * 16;

  // Async global -> LDS stage of the A tile (16 B per lane per issue).
  {
    const unsigned long long gbase =
        (unsigned long long)(uintptr_t)(feat + (size_t)row0 * DIM);
    const unsigned ldsbase = (unsigned)(uintptr_t)ldsA;  // LDS offset = addr[31:0]
    #pragma unroll
    for (int it = 0; it < 16; ++it) {                    // 2048 b128 total
      const int t = tid + it * 128;
      unsigned           lo = ldsbase + (unsigned)t * 16u;
      unsigned long long ga = gbase + (unsigned long long)t * 16ull;
      asm volatile("global_load_async_to_lds_b128 %0, %1, off"
                   :: "v"(lo), "v"(ga) : "memory");
    }
    asm volatile("s_wait_asynccnt 0" ::: "memory");
  }
  if (tid < 16) ldsAvg[tid] = avg[row0 + tid];
  __syncthreads();

  const int wave = tid >> 5;
  const int lane = tid & 31;
  const int hi   = lane >> 4;        // half-wave selector (K pair)
  const int lm   = lane & 15;
  const int n    = wave * 16 + lm;   // this lane's hidden-unit column

  v8f acc = {};
  const float* __restrict__ arow = &ldsA[lm * DIM];
  const v2f*   __restrict__ wp   = (const v2f*)Wp + (hi * HID + n);
  #pragma unroll 4
  for (int k = 0; k < DIM; k += 4) {
    v2f a  = *(const v2f*)(arow + k + 2 * hi);  // ds_load b64 (A fragment)
    v2f bb = wp[(k >> 1) * HID];                // global b64 (B fragment)
    acc = __builtin_amdgcn_wmma_f32_16x16x4_f32(
        /*neg_a=*/false, a, /*neg_b=*/false, bb,
        /*c_mod=*/(short)0, acc, /*reuse_a=*/false, /*reuse_b=*/false);
  }

  // Fold in the avg_dist column (row 512 of W1), bias, ReLU; spill to LDS.
  const float w1last = W1[DIM * HID + n];
  const float bias1  = b1[n];
  #pragma unroll
  for (int r = 0; r < 8; ++r) {
    float v = acc[r] + ldsAvg[r + 8 * hi] * w1last + bias1;
    v = fmaxf(v, 0.0f);
    ldsH[(r + 8 * hi) * HID + n] = v;
  }
  __syncthreads();

  // Layer 2: 64-wide dot + sigmoid, one thread per row.
  if (tid < 16) {
    float s = b2[0];
    #pragma unroll 8
    for (int j = 0; j < HID; ++j) s = fmaf(ldsH[tid * HID + j], W2[j], s);
    out[row0 + tid] = 1.0f / (1.0f + __expf(-s));
  }
}

// ---------------------------------------------------------------------------
extern "C" void kernel_launch(void* const* d_in, const int* in_sizes, int n_in,
                              void* d_out, int out_size, void* d_ws, size_t ws_size,
                              hipStream_t stream) {
  const float* feat  = (const float*)d_in[0];  // spatial_features [B,512]
  const float* cur   = (const float*)d_in[1];  // current_coords   [B,2]
  const float* train = (const float*)d_in[2];  // training_coords  [N,2]
  const float* W1    = (const float*)d_in[3];  // [513,64]
  const float* b1    = (const float*)d_in[4];  // [64]
  const float* W2    = (const float*)d_in[5];  // [64,1]
  const float* b2    = (const float*)d_in[6];  // [1]
  float* out = (float*)d_out;                  // [B,1]

  const int B = out_size;            // 4096
  const int N = in_sizes[2] / 2;     // 100000

  // Workspace layout: [avg: B floats][Wp: 512*64 floats, 256B-aligned]
  float* avg = (float*)d_ws;
  float* Wp  = (float*)((char*)d_ws + (((size_t)B * 4 + 255) & ~(size_t)255));

  pack_w1_kernel<<<(DIM * HID) / 256, 256, 0, stream>>>(W1, Wp);
  topk_avg_kernel<<<B / 2, 32, 0, stream>>>(cur, train, avg, N);
  mlp_wmma_kernel<<<B / 16, 128, 0, stream>>>(feat, avg, Wp, W1, b1, W2, b2, out);
}